// LSTM_Modified_69054484185382
// MI455X (gfx1250) — compile-verified
//
#include <hip/hip_runtime.h>

// Problem constants (from reference): B=64, T=512, IN=128, H=512, OUT=128, 2 layers
#define BATCH  64
#define TSTEPS 512
#define INDIM  128
#define HDIM   512
#define OUTDIM 128

// K-chunk staged through LDS per double-buffer slot (in floats along K)
#define KC 64
// Fragments per chunk: (KC/4) k-steps x 4 gates x 32 lanes x v2f = 4096 floats (16 KB)
#define CHUNK_FLOATS (16 * 4 * 32 * 2)

typedef __attribute__((ext_vector_type(2))) float v2f;
typedef __attribute__((ext_vector_type(8))) float v8f;

__device__ __forceinline__ float sigmoidf_(float x) {
    return 1.0f / (1.0f + __expf(-x));
}

// Block = 128 threads (4 waves), one block per 16-wide hidden tile j0.
// Wave w owns batch tile b0 = 16*w; all 4 waves SHARE the 64 weight rows
// (4 gates x 16 hidden cols), staged global->LDS with CDNA5 async-to-LDS
// copies in B-fragment order, double buffered.
//
// WMMA fp32 16x16x4 fragment layouts (ISA 7.12.2):
//   A (16x4):  lane L holds row M=L%16; regs {K=2*(L/16), K=2*(L/16)+1}
//   B ( 4x16): lane L holds col N=L%16; regs {K=2*(L/16), K=2*(L/16)+1}
//   C/D:       reg r of lane L is element (M = r + 8*(L/16), N = L%16)
__global__ __launch_bounds__(128) void lstm_step_kernel(
    const float* __restrict__ xin, int xin_stride, int K1,
    const float* __restrict__ W_ih,   // (4H x K1) row-major
    const float* __restrict__ W_hh,   // (4H x H)  row-major
    const float* __restrict__ b_ih,   // (4H)
    const float* __restrict__ b_hh,   // (4H)
    const float* __restrict__ h_in,   // (B x H)
    float* __restrict__ h_out,        // (B x H)
    float* __restrict__ c_state)      // (B x H), tile-exclusive RMW
{
    __shared__ float smem[2 * CHUNK_FLOATS];   // 32 KB double buffer

    const int tid  = threadIdx.x;
    const int lane = tid & 31;
    const int wave = tid >> 5;
    const int nloc = lane & 15;
    const int hi   = lane >> 4;
    const int j0   = blockIdx.x * 16;   // hidden tile
    const int b0   = wave * 16;         // batch tile (per wave)

    const int nch0 = K1 / KC;           // chunks from W_ih / xin
    const int nch  = nch0 + HDIM / KC;  // + chunks from W_hh / h_in

    // LDS byte offset of smem: low 32 bits of the flat shared-aperture address
    // are the LDS offset (ISA 10.2).
    const unsigned smem_base = (unsigned)(uintptr_t)&smem[0];

    // Issue async global->LDS copies of one weight chunk, pre-swizzled into
    // B-fragment order: pair p -> (k4 = p>>7, g = (p>>5)&3, lane = p&31).
    auto stage = [&](int buf, int ci) {
        const float* Wsrc;
        int Ksrc, K0;
        if (ci < nch0) { Wsrc = W_ih; Ksrc = K1;   K0 = ci * KC; }
        else           { Wsrc = W_hh; Ksrc = HDIM; K0 = (ci - nch0) * KC; }
        const unsigned base = smem_base + (unsigned)(buf * CHUNK_FLOATS * 4);
#pragma unroll
        for (int jj = 0; jj < 16; ++jj) {          // 2048 v2f pairs / 128 threads
            const int p  = jj * 128 + tid;
            const int L  = p & 31;
            const int q  = p >> 5;                  // k4*4 + g
            const int g  = q & 3;
            const int k4 = q >> 2;
            const int nn = L & 15;
            const int hh = L >> 4;
            const float* gp = Wsrc + (size_t)(g * HDIM + j0 + nn) * Ksrc
                                   + K0 + k4 * 4 + 2 * hh;
            const unsigned ldsoff = base + (unsigned)(p * 8);
            asm volatile("global_load_async_to_lds_b64 %0, %1, off"
                         :: "v"(ldsoff), "v"(gp) : "memory");
        }
    };

    // Accumulators, biases folded in
    v8f acc[4];
#pragma unroll
    for (int g = 0; g < 4; ++g) {
        const float bv = b_ih[g * HDIM + j0 + nloc] + b_hh[g * HDIM + j0 + nloc];
#pragma unroll
        for (int r = 0; r < 8; ++r) acc[g][r] = bv;
    }

    stage(0, 0);
    for (int ci = 0; ci < nch; ++ci) {
        const int buf = ci & 1;
        if (ci + 1 < nch) {
            stage(buf ^ 1, ci + 1);
            // each stage() = 16 async instrs per wave; in-order completion =>
            // <=16 outstanding means the CURRENT chunk is fully in LDS
            asm volatile("s_wait_asynccnt 0x10" ::: "memory");
        } else {
            asm volatile("s_wait_asynccnt 0x0" ::: "memory");
        }
        __syncthreads();

        // A-fragments for this chunk -> registers (one global wait, then LDS-only)
        const float* Asrc;
        int Astr, K0;
        if (ci < nch0) { Asrc = xin;  Astr = xin_stride; K0 = ci * KC; }
        else           { Asrc = h_in; Astr = HDIM;       K0 = (ci - nch0) * KC; }
        const float* arow = Asrc + (size_t)(b0 + nloc) * Astr + K0 + 2 * hi;
        v2f areg[16];
#pragma unroll
        for (int k4 = 0; k4 < 16; ++k4) {
            areg[k4].x = arow[k4 * 4];
            areg[k4].y = arow[k4 * 4 + 1];
        }

        const v2f* bfrag = (const v2f*)&smem[buf * CHUNK_FLOATS];
#pragma unroll
        for (int k4 = 0; k4 < 16; ++k4) {
#pragma unroll
            for (int g = 0; g < 4; ++g) {
                const v2f b = bfrag[(k4 * 4 + g) * 32 + lane];   // ds_load_b64
                acc[g] = __builtin_amdgcn_wmma_f32_16x16x4_f32(
                    false, areg[k4], false, b, (short)0, acc[g], false, false);
            }
        }
        __syncthreads();   // buffer reused by stage() two chunks later
    }

    // ---- elementwise LSTM cell on this wave's 16x16 tile ----
#pragma unroll
    for (int r = 0; r < 8; ++r) {
        const int m = b0 + r + 8 * hi;
        const size_t idx = (size_t)m * HDIM + j0 + nloc;
        const float cold = c_state[idx];
        const float iv = sigmoidf_(acc[0][r]);
        const float fv = sigmoidf_(acc[1][r]);
        const float gv = tanhf(acc[2][r]);
        const float ov = sigmoidf_(acc[3][r]);
        const float cn = fv * cold + iv * gv;
        c_state[idx] = cn;
        h_out[idx]   = ov * tanhf(cn);
    }
}

// out(64x128) = last(64x512) @ W_fc(128x512)^T + b_fc, one wave per 16x16 tile.
__global__ __launch_bounds__(32) void fc_kernel(
    const float* __restrict__ last, const float* __restrict__ W_fc,
    const float* __restrict__ b_fc, float* __restrict__ out)
{
    const int lane = threadIdx.x;
    const int nloc = lane & 15;
    const int hi   = lane >> 4;
    const int b0   = blockIdx.x * 16;
    const int n0   = blockIdx.y * 16;

    v8f acc;
    const float bv = b_fc[n0 + nloc];
#pragma unroll
    for (int r = 0; r < 8; ++r) acc[r] = bv;

    const float* arow = last + (size_t)(b0 + nloc) * HDIM + 2 * hi;
    const float* wrow = W_fc + (size_t)(n0 + nloc) * HDIM + 2 * hi;
#pragma unroll 4
    for (int k = 0; k < HDIM; k += 4) {
        v2f a; a.x = arow[k]; a.y = arow[k + 1];
        v2f b; b.x = wrow[k]; b.y = wrow[k + 1];
        acc = __builtin_amdgcn_wmma_f32_16x16x4_f32(false, a, false, b, (short)0, acc, false, false);
    }

#pragma unroll
    for (int r = 0; r < 8; ++r)
        out[(size_t)(b0 + r + 8 * hi) * OUTDIM + n0 + nloc] = acc[r];
}

extern "C" void kernel_launch(void* const* d_in, const int* in_sizes, int n_in,
                              void* d_out, int out_size, void* d_ws, size_t ws_size,
                              hipStream_t stream) {
    (void)in_sizes; (void)n_in; (void)out_size; (void)ws_size;

    const float* x     = (const float*)d_in[0];
    const float* W_ih0 = (const float*)d_in[1];
    const float* W_hh0 = (const float*)d_in[2];
    const float* b_ih0 = (const float*)d_in[3];
    const float* b_hh0 = (const float*)d_in[4];
    const float* W_ih1 = (const float*)d_in[5];
    const float* W_hh1 = (const float*)d_in[6];
    const float* b_ih1 = (const float*)d_in[7];
    const float* b_hh1 = (const float*)d_in[8];
    const float* W_fc  = (const float*)d_in[9];
    const float* b_fc  = (const float*)d_in[10];
    float* out = (float*)d_out;

    const size_t BH = (size_t)BATCH * HDIM;   // 32768 floats
    float* h0a = (float*)d_ws;
    float* h0b = h0a + BH;
    float* c0  = h0b + BH;
    float* h1a = c0  + BH;
    float* h1b = h1a + BH;
    float* c1  = h1b + BH;

    // zero-init h/c state (d_ws is poisoned by the harness); capture-legal
    hipMemsetAsync(d_ws, 0, 6 * BH * sizeof(float), stream);

    const dim3 blk(128);                 // 4 waves: batch tiles 0..3
    const dim3 grid(HDIM / 16);          // 32 blocks: hidden tiles

    for (int t = 0; t < TSTEPS; ++t) {
        float* h0_in  = (t & 1) ? h0b : h0a;
        float* h0_out = (t & 1) ? h0a : h0b;
        float* h1_in  = (t & 1) ? h1b : h1a;
        float* h1_out = (t & 1) ? h1a : h1b;

        // layer 0: input x[:, t, :]  (row stride T*IN between batch rows)
        lstm_step_kernel<<<grid, blk, 0, stream>>>(
            x + (size_t)t * INDIM, TSTEPS * INDIM, INDIM,
            W_ih0, W_hh0, b_ih0, b_hh0, h0_in, h0_out, c0);

        // layer 1: input is layer-0 hidden state at step t
        lstm_step_kernel<<<grid, blk, 0, stream>>>(
            h0_out, HDIM, HDIM,
            W_ih1, W_hh1, b_ih1, b_hh1, h1_in, h1_out, c1);
    }

    // t = 511 (odd) wrote h1_out = h1a
    float* hlast = h1a;
    const dim3 gfc(BATCH / 16, OUTDIM / 16);  // 4 x 8 waves
    fc_kernel<<<gfc, blk.x == 128 ? dim3(32) : dim3(32), 0, stream>>>(hlast, W_fc, b_fc, out);
}